// MXQuantizer_21363167330407
// MI455X (gfx1250) — compile-verified
//
#include <hip/hip_runtime.h>
#include <stdint.h>

// MX-FP8 (e4m3, block=32) quantize-dequantize of an 8192x8192 f32 tensor.
// Memory-bound: 512 MB traffic, ~22 us floor at 23.3 TB/s. wave32 mapping:
// 8 lanes x float4 = one 32-element MX block; amax via ds_swizzle xor-reduce.
// 4 float4s per thread for MLP + preamble amortization. Input staged through
// LDS with GLOBAL_LOAD_ASYNC_TO_LDS_B128 (ASYNCcnt-tracked); stores are
// nontemporal b128.

typedef float vf4 __attribute__((ext_vector_type(4)));

#if defined(__gfx1250__) && __has_builtin(__builtin_amdgcn_global_load_async_to_lds_b128)
#define HAS_ASYNC 1
#else
#define HAS_ASYNC 0
#endif

// The async builtin's parameters are GCC-vector int4 pointers in AS(1)/AS(3)
// (per hipcc diagnostic: "__attribute__((__vector_size__(4*sizeof(int)))) int __device__ *").
typedef int v4i_g __attribute__((vector_size(16)));
typedef __attribute__((address_space(1))) v4i_g global_v4i;
typedef __attribute__((address_space(3))) v4i_g lds_v4i;

template <unsigned short N>
__device__ __forceinline__ void async_wait() {
#if defined(__gfx1250__)
#if __has_builtin(__builtin_amdgcn_s_wait_asynccnt)
    __builtin_amdgcn_s_wait_asynccnt(N);
#else
    asm volatile("s_wait_asynccnt %0" ::"i"(N) : "memory");
#endif
#endif
}

// Round f32 -> bf16 (RNE), returned as f32. Emulating each bf16 add/mul as
// exact f32 compute + RNE re-round is bit-identical to native bf16 RNE.
__device__ __forceinline__ float bf16_rne(float f) {
    uint32_t u = __float_as_uint(f);
    u += 0x7FFFu + ((u >> 16) & 1u);
    return __uint_as_float(u & 0xFFFF0000u);
}

// XOR-lane swizzle on the LDS datapath (no LDS memory, no barrier).
// offset encoding (group-of-32): [14:10]=xor_mask, [9:5]=or_mask, [4:0]=and_mask
#define SWZ_XOR(v, imm) \
    __uint_as_float((uint32_t)__builtin_amdgcn_ds_swizzle((int)__float_as_uint(v), (imm)))

// One element of the quantize-dequantize round trip.
// private_exp = max(floor(log2|xs|), -6) from the exponent field (field 0 for
// zero/denorm -> -127 -> clamped to -6, matching the reference result).
// round 'nearest' = sign * floor(|t| + 0.5) (half away from zero).
__device__ __forceinline__ float mx_elem(float x, float rcp_s, float s) {
    float xs = x * rcp_s;
    uint32_t b = __float_as_uint(xs);
    int e = (int)((b >> 23) & 0xFFu) - 127;
    e = (e < -6) ? -6 : e;
    float t = ldexpf(fabsf(xs), 3 - e);                // |xs|/2^e * 2^(MBITS-2)
    float r = floorf(t + 0.5f);
    float out = copysignf(ldexpf(r, e - 3), xs);
    out = fminf(fmaxf(out, -448.0f), 448.0f);          // saturate_normals
    if ((b & 0x7F800000u) == 0x7F800000u) out = xs;    // Inf/NaN passthrough
    return out * s;                                    // dequantize
}

// Full 4-element (eighth-block) pipeline: block amax via 3 xor-swizzles over
// the owning 8-lane group, bf16 scale with nano-mantissa rounding, then the
// per-element quant-dequant round trip. Two IEEE divisions per 4 elements (scale, reciprocal).
__device__ __forceinline__ vf4 mx_process4(vf4 v) {
    float am = fmaxf(fmaxf(fabsf(v.x), fabsf(v.y)), fmaxf(fabsf(v.z), fabsf(v.w)));
    am = fmaxf(am, SWZ_XOR(am, 0x041F));   // xor 1
    am = fmaxf(am, SWZ_XOR(am, 0x081F));   // xor 2
    am = fmaxf(am, SWZ_XOR(am, 0x101F));   // xor 4 -> block amax in all 8 lanes

    float scale = bf16_rne(am / 448.0f);       // f32 div, RNE cast to bf16
    float t     = scale * 256.0f;              // exact pow2 (bf16-exact)
    float rr    = floorf(bf16_rne(t + 0.5f));  // bf16 add emulated exactly
    float s     = rr * (1.0f / 256.0f);        // exact pow2, bf16-representable
    s = (s == 0.0f) ? 1.0f : s;
    float rcp_s = 1.0f / s;                    // IEEE-exact, amortized over 4

    vf4 o;
    o.x = mx_elem(v.x, rcp_s, s);
    o.y = mx_elem(v.y, rcp_s, s);
    o.z = mx_elem(v.z, rcp_s, s);
    o.w = mx_elem(v.w, rcp_s, s);
    return o;
}

#define MXQ_T 256   // threads per block (8 wave32s)
#define MXQ_K 4     // float4s per thread

__global__ __launch_bounds__(MXQ_T) void mx_quant_kernel(const float* __restrict__ x,
                                                         float* __restrict__ out,
                                                         int n4) {
    const int tid = threadIdx.x;
    const long long base = (long long)blockIdx.x * (MXQ_T * MXQ_K);
    const vf4* __restrict__ xin = (const vf4*)x;
    vf4* __restrict__ xout = (vf4*)out;

    if (base + MXQ_T * MXQ_K <= (long long)n4) {
#if HAS_ASYNC
        // Stage via the gfx1250 async copy engine: 4 in-flight ASYNCcnt ops
        // per wave; staged s_wait_asynccnt releases each slice in order.
        __shared__ vf4 stage[MXQ_K * MXQ_T];
#pragma unroll
        for (int k = 0; k < MXQ_K; ++k) {
            const vf4* gsrc = xin + base + k * MXQ_T + tid;
            global_v4i* g1 = (global_v4i*)(uintptr_t)gsrc;  // flat == global addr
            // generic->LDS: flat addr[31:0] is the LDS offset (ISA aperture rule)
            lds_v4i* l3 = (lds_v4i*)(uint32_t)(uintptr_t)&stage[k * MXQ_T + tid];
            __builtin_amdgcn_global_load_async_to_lds_b128(g1, l3, /*offset=*/0, /*cpol=*/0);
        }
#define MXQ_STAGE(k, w)                                                               \
    do {                                                                              \
        async_wait<w>();                                                              \
        vf4 v = stage[(k) * MXQ_T + tid];                                             \
        __builtin_nontemporal_store(mx_process4(v), xout + base + (k) * MXQ_T + tid); \
    } while (0)
        MXQ_STAGE(0, 3);
        MXQ_STAGE(1, 2);
        MXQ_STAGE(2, 1);
        MXQ_STAGE(3, 0);
#undef MXQ_STAGE
#else
        // Fallback: direct 128-bit nontemporal loads, 4 in flight per wave.
        vf4 v[MXQ_K];
#pragma unroll
        for (int k = 0; k < MXQ_K; ++k)
            v[k] = __builtin_nontemporal_load(xin + base + k * MXQ_T + tid);
#pragma unroll
        for (int k = 0; k < MXQ_K; ++k)
            __builtin_nontemporal_store(mx_process4(v[k]), xout + base + k * MXQ_T + tid);
#endif
    } else {
        // Guarded tail (unused for 8192x8192: 16,777,216 float4s = 16384 full blocks)
        for (int k = 0; k < MXQ_K; ++k) {
            long long i = base + k * MXQ_T + tid;
            if (i < (long long)n4) {
                vf4 v = __builtin_nontemporal_load(xin + i);
                __builtin_nontemporal_store(mx_process4(v), xout + i);
            }
        }
    }
}

extern "C" void kernel_launch(void* const* d_in, const int* in_sizes, int n_in,
                              void* d_out, int out_size, void* d_ws, size_t ws_size,
                              hipStream_t stream) {
    const float* x = (const float*)d_in[0];
    float* out = (float*)d_out;
    int n  = in_sizes[0];   // 8192*8192 = 67,108,864
    int n4 = n >> 2;        // 16,777,216 float4s
    int per_block = MXQ_T * MXQ_K;
    int grid = (n4 + per_block - 1) / per_block;   // 16384 blocks
    mx_quant_kernel<<<grid, MXQ_T, 0, stream>>>(x, out, n4);
}